// FK_Velocity_Loss_64149631533149
// MI455X (gfx1250) — compile-verified
//
#include <hip/hip_runtime.h>
#include <stdint.h>

// ---------------------------------------------------------------------------
// FK velocity loss for MI455X (gfx1250, wave32).
//   chains = N*2, each chain = 64 contiguous f32 (four 4x4 row-major mats).
//   t = (M0*M1*M2*M3)[:3,3] = M0*(M1*(M2*(M3 col3)))  -> 3 matvecs/chain.
//   pos_loss = mean((t_out - t_gt)^2), vel_loss = mean(((t_out-t_prev)-(t_gt-t_prev))^2)
// Memory-bound: 614 MB streamed -> use async global->LDS B128 staging for
// coalescing, WMMA f32_16x16x4 (B = ones) as the wave-level sum reducer.
// ---------------------------------------------------------------------------

typedef __attribute__((ext_vector_type(2))) float v2f;
typedef __attribute__((ext_vector_type(8))) float v8f;

#define CHAINS_PER_BLOCK 256
#define CHAIN_STRIDE_F   68   // 64 payload floats + 4 pad (272B, 16B aligned)

// Stage 256 chains (64KB) of one pose array into LDS, coalesced, via the
// CDNA5 async global->LDS path (tracked by ASYNCcnt).
__device__ __forceinline__ void stage_tile_async(const float4* gsrc,
                                                 unsigned ldsDynBase) {
  #pragma unroll
  for (int i = 0; i < 16; ++i) {
    int f     = (int)threadIdx.x + i * 256;   // float4 index within 64KB tile
    int chain = f >> 4;                       // 16 float4 per chain
    int k     = f & 15;
    unsigned ldsOff = ldsDynBase + (unsigned)(chain * (CHAIN_STRIDE_F * 4) + k * 16);
    uint64_t gaddr  = (uint64_t)(uintptr_t)(gsrc + f);
    asm volatile("global_load_async_to_lds_b128 %0, %1, off"
                 :: "v"(ldsOff), "v"(gaddr) : "memory");
  }
  asm volatile("s_wait_asynccnt 0" ::: "memory");
}

// Translation column of the 4-matrix chain product. cbase -> 64 floats.
__device__ __forceinline__ void fk_translate(const float* cbase, float t[3]) {
  const float4* P = (const float4*)cbase;
  float v0 = P[12].w, v1 = P[13].w, v2 = P[14].w, v3 = P[15].w; // M3[:,3]
  #pragma unroll
  for (int j = 2; j >= 0; --j) {
    float4 r0 = P[j * 4 + 0], r1 = P[j * 4 + 1];
    float4 r2 = P[j * 4 + 2], r3 = P[j * 4 + 3];
    float n0 = fmaf(r0.x, v0, fmaf(r0.y, v1, fmaf(r0.z, v2, r0.w * v3)));
    float n1 = fmaf(r1.x, v0, fmaf(r1.y, v1, fmaf(r1.z, v2, r1.w * v3)));
    float n2 = fmaf(r2.x, v0, fmaf(r2.y, v1, fmaf(r2.z, v2, r2.w * v3)));
    float n3 = fmaf(r3.x, v0, fmaf(r3.y, v1, fmaf(r3.z, v2, r3.w * v3)));
    v0 = n0; v1 = n1; v2 = n2; v3 = n3;
  }
  t[0] = v0; t[1] = v1; t[2] = v2;
}

// Wave32 sum via v_wmma_f32_16x16x4_f32 with B = ones (B layout-independent).
// A layout (documented): lane L holds A[L%16][(L/16)*2 + vgpr]. With
// A = {x, 0} per lane, D row m (all columns) = x(lane m) + x(lane m+16).
// A lane's 8 D regs hold rows {0..7} (lanes 0-15) or {8..15} (lanes 16-31),
// so sum(c[0..7]) + shfl_xor(.,16) = full wave sum in every lane.
__device__ __forceinline__ float wave_sum_wmma(float x) {
  v2f a; a.x = x;    a.y = 0.0f;
  v2f b; b.x = 1.0f; b.y = 1.0f;
  v8f c = {};
  v8f d = __builtin_amdgcn_wmma_f32_16x16x4_f32(
      /*neg_a=*/false, a, /*neg_b=*/false, b,
      /*c_mod=*/(short)0, c, /*reuse_a=*/false, /*reuse_b=*/false);
  float s = d[0] + d[1] + d[2] + d[3] + d[4] + d[5] + d[6] + d[7];
  s += __shfl_xor(s, 16, 32);
  return s;
}

__global__ void __launch_bounds__(256)
fk_loss_main(const float* __restrict__ out_pose,
             const float* __restrict__ gt_pose,
             const float* __restrict__ gt_prev_pose,
             float* __restrict__ acc) {
  extern __shared__ float lds[];
  const unsigned tid        = threadIdx.x;
  const unsigned ldsDynBase = __builtin_amdgcn_groupstaticsize();
  const size_t   tileBase   = (size_t)blockIdx.x * CHAINS_PER_BLOCK * 64; // floats
  const float*   myChain    = &lds[tid * CHAIN_STRIDE_F];

  float to[3], tg[3], tp[3];

  stage_tile_async((const float4*)(out_pose + tileBase), ldsDynBase);
  __syncthreads();
  fk_translate(myChain, to);
  __syncthreads();

  stage_tile_async((const float4*)(gt_pose + tileBase), ldsDynBase);
  __syncthreads();
  fk_translate(myChain, tg);
  __syncthreads();

  stage_tile_async((const float4*)(gt_prev_pose + tileBase), ldsDynBase);
  __syncthreads();
  fk_translate(myChain, tp);

  float pos = 0.0f, vel = 0.0f;
  #pragma unroll
  for (int c = 0; c < 3; ++c) {
    float d  = to[c] - tg[c];
    pos = fmaf(d, d, pos);
    float ov = to[c] - tp[c];
    float gv = tg[c] - tp[c];
    float e  = ov - gv;
    vel = fmaf(e, e, vel);
  }

  // EXEC is all-ones here (no divergence in this kernel) -> WMMA legal.
  float wp = wave_sum_wmma(pos);
  float wv = wave_sum_wmma(vel);
  if ((tid & 31u) == 0u) {
    atomicAdd(&acc[0], wp);
    atomicAdd(&acc[1], wv);
  }
}

// Generic tail (chains not multiple of 256) — plain path, no WMMA/LDS.
__global__ void fk_loss_tail(const float* __restrict__ out_pose,
                             const float* __restrict__ gt_pose,
                             const float* __restrict__ gt_prev_pose,
                             float* __restrict__ acc,
                             long firstChain, long nChains) {
  long chain = firstChain + blockIdx.x * blockDim.x + threadIdx.x;
  if (chain >= nChains) return;
  size_t base = (size_t)chain * 64;
  float to[3], tg[3], tp[3];
  fk_translate(out_pose + base, to);
  fk_translate(gt_pose + base, tg);
  fk_translate(gt_prev_pose + base, tp);
  float pos = 0.0f, vel = 0.0f;
  #pragma unroll
  for (int c = 0; c < 3; ++c) {
    float d  = to[c] - tg[c];
    pos = fmaf(d, d, pos);
    float e  = (to[c] - tp[c]) - (tg[c] - tp[c]);
    vel = fmaf(e, e, vel);
  }
  atomicAdd(&acc[0], pos);
  atomicAdd(&acc[1], vel);
}

__global__ void fk_loss_init(float* __restrict__ acc) {
  if (threadIdx.x < 2) acc[threadIdx.x] = 0.0f;
}

__global__ void fk_loss_finalize(const float* __restrict__ acc,
                                 float* __restrict__ out, float invCount) {
  if (threadIdx.x < 2) out[threadIdx.x] = acc[threadIdx.x] * invCount;
}

extern "C" void kernel_launch(void* const* d_in, const int* in_sizes, int n_in,
                              void* d_out, int out_size, void* d_ws, size_t ws_size,
                              hipStream_t stream) {
  const float* out_pose     = (const float*)d_in[0];
  const float* gt_pose      = (const float*)d_in[1];
  const float* gt_prev_pose = (const float*)d_in[2];
  // d_in[3] (gt_pos) is unused by the reference.

  float* acc = (float*)d_ws;     // acc[0]=pos sum, acc[1]=vel sum
  float* out = (float*)d_out;    // out[0]=pos_loss, out[1]=vel_loss

  const long chains     = (long)in_sizes[0] / 64;           // N*2 (= 800000)
  const int  fullBlocks = (int)(chains / CHAINS_PER_BLOCK); // 3125 for N=400000
  const long tailStart  = (long)fullBlocks * CHAINS_PER_BLOCK;
  const long tailCnt    = chains - tailStart;
  const float invCount  = 1.0f / (float)(chains * 3);
  const size_t shBytes  = (size_t)CHAINS_PER_BLOCK * CHAIN_STRIDE_F * sizeof(float);

  hipLaunchKernelGGL(fk_loss_init, dim3(1), dim3(32), 0, stream, acc);
  if (fullBlocks > 0) {
    hipLaunchKernelGGL(fk_loss_main, dim3(fullBlocks), dim3(256), shBytes, stream,
                       out_pose, gt_pose, gt_prev_pose, acc);
  }
  if (tailCnt > 0) {
    int tb = (int)((tailCnt + 255) / 256);
    hipLaunchKernelGGL(fk_loss_tail, dim3(tb), dim3(256), 0, stream,
                       out_pose, gt_pose, gt_prev_pose, acc, tailStart, chains);
  }
  hipLaunchKernelGGL(fk_loss_finalize, dim3(1), dim3(32), 0, stream,
                     acc, out, invCount);
}